// MetaQDA_MAP_16982300688661
// MI455X (gfx1250) — compile-verified
//
#include <hip/hip_runtime.h>

#define DD 640
#define CC 64
#define NN 1024
#define QQ 2048

typedef float v2f __attribute__((ext_vector_type(2)));
typedef float v8f __attribute__((ext_vector_type(8)));
typedef int v4i_vs __attribute__((vector_size(4 * sizeof(int))));
typedef __attribute__((address_space(1))) v4i_vs* as1_v4i_ptr;
typedef __attribute__((address_space(3))) v4i_vs* as3_v4i_ptr;

__device__ __forceinline__ v8f v8f_zero() {
  v8f r;
  #pragma unroll
  for (int i = 0; i < 8; ++i) r[i] = 0.f;
  return r;
}

// D(16x16,f32) = A(16x4,f32) * B(4x16,f32) + C
__device__ __forceinline__ v8f wmma_f32_16x16x4(v2f a, v2f b, v8f c) {
  return __builtin_amdgcn_wmma_f32_16x16x4_f32(false, a, false, b, (short)0, c,
                                               false, false);
}

// Async global -> LDS copy of 16 bytes per lane (ASYNCcnt-tracked).
__device__ __forceinline__ void async_copy_b128(const float* g, float* l) {
  __builtin_amdgcn_global_load_async_to_lds_b128((as1_v4i_ptr)g, (as3_v4i_ptr)l,
                                                 0, 0);
}

__device__ __forceinline__ void wait_asynccnt0() {
  asm volatile("s_wait_asynccnt 0x0" ::: "memory");
}

// ---------------------------------------------------------------------------
// 1) L = diag(|triu_diag|) + triu_lower * strict_lower_mask
// ---------------------------------------------------------------------------
__global__ void k_buildL(const float* __restrict__ diag,
                         const float* __restrict__ lower,
                         float* __restrict__ L) {
  int lin = blockIdx.x * blockDim.x + threadIdx.x;
  if (lin >= DD * DD) return;
  int i = lin / DD, j = lin - i * DD;
  float v = (i == j) ? fabsf(diag[i]) : (i > j ? lower[lin] : 0.0f);
  L[lin] = v;
}

// ---------------------------------------------------------------------------
// 2) LLt = L * L^T via f32 WMMA. One 16x16 output tile per wave.
// ---------------------------------------------------------------------------
__global__ __launch_bounds__(256) void k_llt(const float* __restrict__ L,
                                             float* __restrict__ LLt) {
  int wave = threadIdx.x >> 5, lane = threadIdx.x & 31;
  int tile = blockIdx.x * 8 + wave;  // 0..1599
  int ti = tile / 40, tj = tile - ti * 40;
  int ml = lane & 15, hi = lane >> 4;

  v8f acc = v8f_zero();
  const float* ra = L + (size_t)(ti * 16 + ml) * DD;
  const float* rb = L + (size_t)(tj * 16 + ml) * DD;
  for (int k = 0; k < DD; k += 4) {
    v2f a = *(const v2f*)(ra + k + 2 * hi);
    v2f b = *(const v2f*)(rb + k + 2 * hi);
    acc = wmma_f32_16x16x4(a, b, acc);
  }
  #pragma unroll
  for (int v = 0; v < 8; ++v)
    LLt[(size_t)(ti * 16 + v + 8 * hi) * DD + tj * 16 + ml] = acc[v];
}

// ---------------------------------------------------------------------------
// 3) Deterministic per-class index lists. Thread c scans n ascending.
// ---------------------------------------------------------------------------
__global__ void k_classify(const int* __restrict__ y, int* __restrict__ Nc,
                           int* __restrict__ idx) {
  int c = threadIdx.x;
  if (c >= CC) return;
  int slot = 0;
  for (int n = 0; n < NN; ++n)
    if (y[n] == c) idx[c * NN + (slot++)] = n;
  Nc[c] = slot;
}

// ---------------------------------------------------------------------------
// 4) mu[c] = (1-w) m + w * mean_c,   w = Nc/(kappa_+Nc)
// ---------------------------------------------------------------------------
__global__ void k_mu(const float* __restrict__ X, const float* __restrict__ m,
                     const float* __restrict__ kappa, const int* __restrict__ Nc,
                     const int* __restrict__ idx, float* __restrict__ mu) {
  int c = blockIdx.x;
  int nc = Nc[c];
  float kap = fabsf(kappa[0]) + 1e-6f;
  float w = (float)nc / (kap + (float)nc);
  const int* id = idx + c * NN;
  for (int d = threadIdx.x; d < DD; d += blockDim.x) {
    float s = 0.f;
    for (int t = 0; t < nc; ++t) s += X[(size_t)id[t] * DD + d];
    float mean = (nc > 0) ? s / (float)nc : 0.f;
    mu[c * DD + d] = (1.f - w) * m[d] + w * mean;
  }
}

// ---------------------------------------------------------------------------
// 5) sigma_c[d,e] = (LLt + S_c + kap*m m^T - (kap+Nc) mu mu^T) / scale
// ---------------------------------------------------------------------------
__global__ __launch_bounds__(256) void k_sigma(
    const float* __restrict__ X, const float* __restrict__ m,
    const float* __restrict__ kappa, const float* __restrict__ nu,
    const int* __restrict__ Nc, const int* __restrict__ idx,
    const float* __restrict__ mu, const float* __restrict__ LLt,
    float* __restrict__ sigma) {
  int c = blockIdx.x / 1600;
  int lin = (blockIdx.x % 1600) * 256 + threadIdx.x;
  int d = lin / DD, e = lin - d * DD;
  int nc = Nc[c];
  float kap = fabsf(kappa[0]) + 1e-6f;
  float nu_ = fmaxf(nu[0], (float)(DD - 1) + 1e-6f);
  const int* id = idx + c * NN;
  float s = 0.f;
  for (int t = 0; t < nc; ++t) {
    const float* xr = X + (size_t)id[t] * DD;
    s += xr[d] * xr[e];
  }
  float mud = mu[c * DD + d], mue = mu[c * DD + e];
  float v = LLt[lin] + s + kap * m[d] * m[e] - (kap + (float)nc) * mud * mue;
  sigma[(size_t)c * DD * DD + lin] = v / (nu_ + (float)nc + (float)DD + 2.f);
}

// ---------------------------------------------------------------------------
// 6) In-place Gauss-Jordan inversion (SPD, no pivoting). One 1024-thread
//    workgroup per class; the 64 x 1.6MB matrices stay L2-resident (105 MB
//    total < 192 MB L2).
// ---------------------------------------------------------------------------
__global__ __launch_bounds__(1024) void k_gj(float* __restrict__ sigma) {
  float* A = sigma + (size_t)blockIdx.x * DD * DD;
  __shared__ float rowk[DD];
  __shared__ float fcol[DD];
  __shared__ float s_pinv;
  const int tid = threadIdx.x;
  for (int k = 0; k < DD; ++k) {
    if (tid == 0) s_pinv = 1.0f / A[(size_t)k * DD + k];
    __syncthreads();
    float pinv = s_pinv;
    for (int j = tid; j < DD; j += 1024)
      rowk[j] = (j == k ? 1.0f : A[(size_t)k * DD + j]) * pinv;
    for (int i = tid; i < DD; i += 1024)
      fcol[i] = A[(size_t)i * DD + k];
    __syncthreads();
    for (int lin = tid; lin < DD * DD; lin += 1024) {
      int i = lin / DD, j = lin - i * DD;
      if (i == k) {
        A[lin] = rowk[j];
      } else {
        float v = (j == k) ? 0.f : A[lin];
        A[lin] = v - fcol[i] * rowk[j];
      }
    }
    __syncthreads();
  }
}

// ---------------------------------------------------------------------------
// 7) h_c = 0.7 * (Sinv_c mu_c) + 0.3 * mu_c ;  const_c = mu^T (0.7 Sinv+0.3 I) mu
// ---------------------------------------------------------------------------
__global__ __launch_bounds__(256) void k_postinv(const float* __restrict__ SIg,
                                                 const float* __restrict__ mu,
                                                 float* __restrict__ hc,
                                                 float* __restrict__ constc) {
  int c = blockIdx.x;
  const float* SI = SIg + (size_t)c * DD * DD;
  __shared__ float smu[DD];
  for (int d = threadIdx.x; d < DD; d += 256) smu[d] = mu[c * DD + d];
  __syncthreads();
  float part = 0.f;
  for (int d = threadIdx.x; d < DD; d += 256) {
    float s = 0.f;
    const float* row = SI + (size_t)d * DD;
    for (int e = 0; e < DD; ++e) s += row[e] * smu[e];
    hc[c * DD + d] = 0.7f * s + 0.3f * smu[d];
    part += smu[d] * (0.7f * s + 0.3f * smu[d]);
  }
  __shared__ float red[256];
  red[threadIdx.x] = part;
  __syncthreads();
  for (int off = 128; off > 0; off >>= 1) {
    if (threadIdx.x < off) red[threadIdx.x] += red[threadIdx.x + off];
    __syncthreads();
  }
  if (threadIdx.x == 0) constc[c] = red[0];
}

// ---------------------------------------------------------------------------
// 8) Main kernel: block = (class c, 64-query tile), 8 wave32s.
//    The Xq tile is staged into LDS in 64x80 k-chunks via async global->LDS
//    b128 copies (double-buffered, ASYNCcnt), so A-fragments come from LDS
//    (pitch 84 floats -> 16 lanes hit 16 distinct banks) while B rows stream
//    from the L2-resident Sinv as coalesced 64B segments.
//    out[q,c] = -(0.7*t1 + 0.3*||x||^2 - 2*x.h_c + const_c)
// ---------------------------------------------------------------------------
#define KCH 80               // k-chunk width
#define APITCH 84            // LDS row pitch in floats (bank-conflict-free)

__global__ __launch_bounds__(256) void k_dist(const float* __restrict__ Xq,
                                              const float* __restrict__ SIg,
                                              const float* __restrict__ hc,
                                              const float* __restrict__ constc,
                                              float* __restrict__ out) {
  int c = blockIdx.x;
  int q0 = blockIdx.y * 64;
  int wave = threadIdx.x >> 5, lane = threadIdx.x & 31;
  int ml = lane & 15, hi = lane >> 4;
  int e0 = wave * 80;
  const float* B = SIg + (size_t)c * DD * DD;

  __shared__ float tA[2][64 * APITCH];

  // 64 rows x 80 floats per chunk = 1280 float4s; 5 per thread, coalesced.
  auto load_chunk = [&](int ch, int buf) {
    const float* src = Xq + (size_t)q0 * DD + ch * KCH;
    float* dst = &tA[buf][0];
    #pragma unroll
    for (int t = 0; t < 5; ++t) {
      int f = threadIdx.x + t * 256;
      int row = f / 20, cg = f - row * 20;
      async_copy_b128(src + (size_t)row * DD + cg * 4, dst + row * APITCH + cg * 4);
    }
  };

  v8f acc[4][5];
  #pragma unroll
  for (int mt = 0; mt < 4; ++mt)
    #pragma unroll
    for (int nt = 0; nt < 5; ++nt) acc[mt][nt] = v8f_zero();

  load_chunk(0, 0);
  int buf = 0;
  for (int ch = 0; ch < DD / KCH; ++ch) {
    wait_asynccnt0();     // my wave's async copies for `buf` are in LDS
    __syncthreads();      // ...and everyone else's too
    if (ch + 1 < DD / KCH) load_chunk(ch + 1, buf ^ 1);

    for (int kk = 0; kk < KCH; kk += 4) {
      v2f a[4];
      #pragma unroll
      for (int mt = 0; mt < 4; ++mt)
        a[mt] = *(const v2f*)&tA[buf][(mt * 16 + ml) * APITCH + kk + 2 * hi];
      v2f b[5];
      int krow = ch * KCH + kk + 2 * hi;
      #pragma unroll
      for (int nt = 0; nt < 5; ++nt) {
        const float* bp = B + (size_t)krow * DD + e0 + nt * 16 + ml;
        b[nt].x = bp[0];
        b[nt].y = bp[DD];
      }
      #pragma unroll
      for (int mt = 0; mt < 4; ++mt)
        #pragma unroll
        for (int nt = 0; nt < 5; ++nt)
          acc[mt][nt] = wmma_f32_16x16x4(a[mt], b[nt], acc[mt][nt]);
    }
    __syncthreads();      // all reads of `buf` done before it is refilled
    buf ^= 1;
  }

  __shared__ float t1part[8][64];
  float* tp = t1part[wave];
  #pragma unroll
  for (int mt = 0; mt < 4; ++mt) {
    #pragma unroll
    for (int v = 0; v < 8; ++v) {
      int mrow = mt * 16 + v + 8 * hi;
      const float* xr = Xq + (size_t)(q0 + mrow) * DD;
      float p = 0.f;
      #pragma unroll
      for (int nt = 0; nt < 5; ++nt)
        p += xr[e0 + nt * 16 + ml] * acc[mt][nt][v];
      // sum over the 16 lanes of this half-wave (same hi, ml = 0..15)
      p += __shfl_xor(p, 1, 32);
      p += __shfl_xor(p, 2, 32);
      p += __shfl_xor(p, 4, 32);
      p += __shfl_xor(p, 8, 32);
      if (ml == 0) tp[mrow] = p;
    }
  }

  __shared__ float xh2[4][64];
  __shared__ float xn2[4][64];
  {
    int qi = threadIdx.x & 63;
    int part = threadIdx.x >> 6;  // 0..3, each covers 160 features
    const float* xr = Xq + (size_t)(q0 + qi) * DD + part * 160;
    const float* h = hc + c * DD + part * 160;
    float xh = 0.f, xn = 0.f;
    for (int d = 0; d < 160; ++d) {
      float x = xr[d];
      xh += x * h[d];
      xn += x * x;
    }
    xh2[part][qi] = xh;
    xn2[part][qi] = xn;
  }
  __syncthreads();

  if (threadIdx.x < 64) {
    int t = threadIdx.x;
    float t1 = 0.f;
    #pragma unroll
    for (int w2 = 0; w2 < 8; ++w2) t1 += t1part[w2][t];
    float xh = xh2[0][t] + xh2[1][t] + xh2[2][t] + xh2[3][t];
    float xn = xn2[0][t] + xn2[1][t] + xn2[2][t] + xn2[3][t];
    out[(size_t)(q0 + t) * CC + c] = -(0.7f * t1 + 0.3f * xn - 2.f * xh + constc[c]);
  }
}

// ---------------------------------------------------------------------------
extern "C" void kernel_launch(void* const* d_in, const int* in_sizes, int n_in,
                              void* d_out, int out_size, void* d_ws,
                              size_t ws_size, hipStream_t stream) {
  (void)in_sizes; (void)n_in; (void)out_size; (void)ws_size;
  const float* X     = (const float*)d_in[0];
  const int*   y     = (const int*)d_in[1];
  const float* Xq    = (const float*)d_in[2];
  const float* m     = (const float*)d_in[3];
  const float* kappa = (const float*)d_in[4];
  const float* nu    = (const float*)d_in[5];
  const float* tdiag = (const float*)d_in[6];
  const float* tlow  = (const float*)d_in[7];
  float* out = (float*)d_out;

  char* ws = (char*)d_ws;
  size_t off = 0;
  auto alloc = [&](size_t bytes) -> void* {
    void* p = ws + off;
    off = (off + bytes + 255) & ~(size_t)255;
    return p;
  };
  float* L      = (float*)alloc(sizeof(float) * DD * DD);
  float* LLt    = (float*)alloc(sizeof(float) * DD * DD);
  float* mu     = (float*)alloc(sizeof(float) * CC * DD);
  float* hc     = (float*)alloc(sizeof(float) * CC * DD);
  float* constc = (float*)alloc(sizeof(float) * CC);
  int*   Nc     = (int*)alloc(sizeof(int) * CC);
  int*   idx    = (int*)alloc(sizeof(int) * CC * NN);
  float* sigma  = (float*)alloc(sizeof(float) * (size_t)CC * DD * DD);

  k_buildL<<<(DD * DD + 255) / 256, 256, 0, stream>>>(tdiag, tlow, L);
  k_llt<<<200, 256, 0, stream>>>(L, LLt);
  k_classify<<<1, 64, 0, stream>>>(y, Nc, idx);
  k_mu<<<CC, 256, 0, stream>>>(X, m, kappa, Nc, idx, mu);
  k_sigma<<<CC * 1600, 256, 0, stream>>>(X, m, kappa, nu, Nc, idx, mu, LLt, sigma);
  k_gj<<<CC, 1024, 0, stream>>>(sigma);
  k_postinv<<<CC, 256, 0, stream>>>(sigma, mu, hc, constc);
  k_dist<<<dim3(CC, QQ / 64), 256, 0, stream>>>(Xq, sigma, hc, constc, out);
}